// Joiner_43542378446962
// MI455X (gfx1250) — compile-verified
//
#include <hip/hip_runtime.h>
#include <hip/hip_bf16.h>

// RNN-T Joiner: out[b,t,u,v] = (enc[b,t,:] + pred[b,u,:]) . W[v,:] + bias[v]
// Decomposition: out = E[b,t,v] + P[b,u,v],
//   E = enc @ W^T + bias   (1600 x 1024)
//   P = pred @ W^T         ( 400 x 1024)
// Phase 1: tiny WMMA f32 GEMM into workspace EP[2000][1024].
// Phase 2: bandwidth-bound broadcast add with NT stores (327.7 MB output).

#define Bsz 8
#define Tsz 200
#define Usz 50
#define Dsz 512
#define Vsz 1024

#define ENC_ROWS (Bsz * Tsz)          // 1600
#define PRED_ROWS (Bsz * Usz)         // 400
#define EP_ROWS (ENC_ROWS + PRED_ROWS)// 2000
#define M_TILES (EP_ROWS / 16)        // 125
#define N_TILES (Vsz / 16)            // 64
#define TOTAL_TILES (M_TILES * N_TILES) // 8000

typedef float v2f __attribute__((ext_vector_type(2)));
typedef float v4f __attribute__((ext_vector_type(4)));
typedef float v8f __attribute__((ext_vector_type(8)));

// ---------------------------------------------------------------------------
// Phase 1: EP[0:1600]   = enc @ W^T + bias
//          EP[1600:2000]= pred @ W^T
// One wave32 per 16x16 tile, V_WMMA_F32_16X16X4_F32 over K=512 (128 steps).
// ---------------------------------------------------------------------------
__global__ __launch_bounds__(256) void joiner_gemm_wmma(
    const float* __restrict__ enc, const float* __restrict__ pred,
    const float* __restrict__ W, const float* __restrict__ bias,
    float* __restrict__ EP)
{
    const int lane = threadIdx.x & 31;
    const int wave = threadIdx.x >> 5;
    const int tile = blockIdx.x * 8 + wave;        // 0 .. 7999 (exact)

    const int mTile = tile % M_TILES;
    const int nTile = tile / M_TILES;
    const int m0 = mTile * 16;
    const int n0 = nTile * 16;

    const int lm   = lane & 15;                    // M index (A) / N index (B)
    const int hi   = lane >> 4;                    // lane half
    const int koff = hi * 2;                       // K offset for upper half

    // A row pointer: tiles are purely enc or purely pred (1600 % 16 == 0),
    // so this select is wave-uniform in practice.
    const int row = m0 + lm;
    const float* __restrict__ arow =
        (row < ENC_ROWS) ? (enc + (size_t)row * Dsz)
                         : (pred + (size_t)(row - ENC_ROWS) * Dsz);
    // B[k][n] = W[n0+n][k]  (W is [V, D] row-major)
    const float* __restrict__ brow = W + (size_t)(n0 + lm) * Dsz;

    v8f c = {0.f, 0.f, 0.f, 0.f, 0.f, 0.f, 0.f, 0.f};

    #pragma unroll 8
    for (int k0 = 0; k0 < Dsz; k0 += 4) {
        // A 16x4 layout: VGPR0/1 = K(0,1) for lanes 0-15, K(2,3) for lanes 16-31
        v2f a = *(const v2f*)(arow + k0 + koff);
        // B 4x16 layout mirrors A with lanes indexing N
        v2f b = *(const v2f*)(brow + k0 + koff);
        // (neg_a, A, neg_b, B, c_mod, C, reuse_a, reuse_b)
        c = __builtin_amdgcn_wmma_f32_16x16x4_f32(
                false, a, false, b, (short)0, c, false, false);
    }

    // Fold bias into enc rows only (depends only on column N = lane&15)
    const float bv = (m0 < ENC_ROWS) ? bias[n0 + lm] : 0.0f;

    // C/D layout: VGPR j -> row M = j + 8*hi, col N = lane&15
    float* __restrict__ outBase =
        EP + (size_t)(m0 + hi * 8) * Vsz + (size_t)(n0 + lm);
    #pragma unroll
    for (int j = 0; j < 8; ++j) {
        outBase[(size_t)j * Vsz] = c[j] + bv;
    }
}

// ---------------------------------------------------------------------------
// Phase 2: out[b,t,u,v] = E[b*T+t][v] + P[b*U+u][v]
// One float4 per thread; EP stays hot in L2 (8.2 MB), output streamed with
// non-temporal stores (327.7 MB >> L2).
// ---------------------------------------------------------------------------
__global__ __launch_bounds__(256) void joiner_bcast(
    const float* __restrict__ EP, float* __restrict__ out)
{
    const int idx = blockIdx.x * 256 + threadIdx.x;   // 0 .. 20,479,999
    const int v4  = idx & (Vsz / 4 - 1);              // V/4 = 256 (pow2)
    const int rem = idx >> 8;                         // (b*T + t)*U + u
    const int u   = rem % Usz;
    const int bt  = rem / Usz;                        // b*T + t  (0..1599)
    const int b   = bt / Tsz;

    const v4f e = *(const v4f*)(EP + (size_t)bt * Vsz + v4 * 4);
    const v4f p = *(const v4f*)(EP + (size_t)(ENC_ROWS + b * Usz + u) * Vsz + v4 * 4);
    const v4f o = e + p;

    __builtin_nontemporal_store(o, reinterpret_cast<v4f*>(out) + idx);
}

// ---------------------------------------------------------------------------
// Fallback (only if workspace is too small): direct per-element dot product.
// ---------------------------------------------------------------------------
__global__ __launch_bounds__(256) void joiner_fallback(
    const float* __restrict__ enc, const float* __restrict__ pred,
    const float* __restrict__ W, const float* __restrict__ bias,
    float* __restrict__ out)
{
    const long long gid = (long long)blockIdx.x * 256 + threadIdx.x;
    if (gid >= (long long)Bsz * Tsz * Usz * Vsz) return;
    const int v = (int)(gid & (Vsz - 1));
    long long r = gid >> 10;             // V = 1024
    const int u  = (int)(r % Usz);
    const long long bt = r / Usz;
    const int b  = (int)(bt / Tsz);

    const float* e = enc  + bt * Dsz;
    const float* p = pred + ((long long)b * Usz + u) * Dsz;
    const float* w = W    + (long long)v * Dsz;
    float acc = bias[v];
    for (int k = 0; k < Dsz; ++k) acc += (e[k] + p[k]) * w[k];
    out[gid] = acc;
}

extern "C" void kernel_launch(void* const* d_in, const int* in_sizes, int n_in,
                              void* d_out, int out_size, void* d_ws, size_t ws_size,
                              hipStream_t stream) {
    const float* enc  = (const float*)d_in[0];  // [B,T,D]
    const float* pred = (const float*)d_in[1];  // [B,U,D]
    const float* W    = (const float*)d_in[2];  // [V,D]
    const float* bias = (const float*)d_in[3];  // [V]
    float* out = (float*)d_out;                 // [B,T,U,V]

    const size_t need = (size_t)EP_ROWS * Vsz * sizeof(float); // 8.192 MB
    if (ws_size >= need) {
        float* EP = (float*)d_ws;
        joiner_gemm_wmma<<<TOTAL_TILES / 8, 256, 0, stream>>>(enc, pred, W, bias, EP);
        const int total4 = Bsz * Tsz * Usz * (Vsz / 4);        // 20,480,000
        joiner_bcast<<<total4 / 256, 256, 0, stream>>>(EP, out);
    } else {
        const long long total = (long long)Bsz * Tsz * Usz * Vsz;
        joiner_fallback<<<(int)((total + 255) / 256), 256, 0, stream>>>(
            enc, pred, W, bias, out);
    }
}